// GauntTensorProductFourier2D_2525440770529
// MI455X (gfx1250) — compile-verified
//
#include <hip/hip_runtime.h>
#include <hip/hip_bf16.h>
#include <math.h>

typedef float v2f __attribute__((ext_vector_type(2)));
typedef float v8f __attribute__((ext_vector_type(8)));

#define NGRIDS 43           // 9 (y1) + 9 (y2) + 25 (z)
#define GRID_SZ 1024        // 32 x 32
#define KPAD 84             // 81 padded to multiple of 4 (21 chunks of 4)
#define OPAD 32             // 25 padded to 32

// ---------------------------------------------------------------------------
// Stage 1: DFT along P axis.  tmp[g][t][k2] = sum_p in[g][t][p] * e^{-2pi i k2 p/32}
// ---------------------------------------------------------------------------
__global__ __launch_bounds__(256) void dft_stage_p(const float2* __restrict__ y1,
                                                   const float2* __restrict__ y2,
                                                   const float2* __restrict__ zg,
                                                   float2* __restrict__ tmp) {
    __shared__ float2 tw[32];
    int tid = threadIdx.x;
    if (tid < 32) {
        float ang = -6.28318530717958647692f * (float)tid / 32.0f;
        float s, c;
        __sincosf(ang, &s, &c);
        tw[tid] = make_float2(c, s);
    }
    __syncthreads();
    int idx = blockIdx.x * 256 + tid;          // (g, t, k2)
    if (idx >= NGRIDS * GRID_SZ) return;
    int g  = idx >> 10;
    int t  = (idx >> 5) & 31;
    int k2 = idx & 31;
    const float2* src = (g < 9)  ? (y1 + g * GRID_SZ)
                      : (g < 18) ? (y2 + (g - 9) * GRID_SZ)
                                 : (zg + (g - 18) * GRID_SZ);
    float re = 0.f, im = 0.f;
    for (int p = 0; p < 32; ++p) {
        float2 v = src[t * 32 + p];
        float2 w = tw[(k2 * p) & 31];
        re += v.x * w.x - v.y * w.y;
        im += v.x * w.y + v.y * w.x;
    }
    tmp[idx] = make_float2(re, im);
}

// ---------------------------------------------------------------------------
// Stage 2: DFT along T axis.  F[g][k1][k2] = sum_t tmp[g][t][k2] * e^{-2pi i k1 t/32}
// ---------------------------------------------------------------------------
__global__ __launch_bounds__(256) void dft_stage_t(const float2* __restrict__ tmp,
                                                   float2* __restrict__ F) {
    __shared__ float2 tw[32];
    int tid = threadIdx.x;
    if (tid < 32) {
        float ang = -6.28318530717958647692f * (float)tid / 32.0f;
        float s, c;
        __sincosf(ang, &s, &c);
        tw[tid] = make_float2(c, s);
    }
    __syncthreads();
    int idx = blockIdx.x * 256 + tid;          // (g, k1, k2)
    if (idx >= NGRIDS * GRID_SZ) return;
    int g  = idx >> 10;
    int k1 = (idx >> 5) & 31;
    int k2 = idx & 31;
    const float2* src = tmp + g * GRID_SZ;
    float re = 0.f, im = 0.f;
    for (int t = 0; t < 32; ++t) {
        float2 v = src[t * 32 + k2];
        float2 w = tw[(k1 * t) & 31];
        re += v.x * w.x - v.y * w.y;
        im += v.x * w.y + v.y * w.x;
    }
    F[idx] = make_float2(re, im);
}

// ---------------------------------------------------------------------------
// Build padded bilinear weight matrix Wt[KPAD][OPAD] (k = a1*9+a2, o = output irrep).
// W = (1/1024) * Re sum_k conj(F1*F2) * Z     (Parseval)
// ---------------------------------------------------------------------------
__global__ __launch_bounds__(256) void build_w(const float2* __restrict__ F,
                                               float* __restrict__ Wt) {
    int idx = blockIdx.x * 256 + threadIdx.x;  // (k, o)
    if (idx >= KPAD * OPAD) return;
    int k = idx >> 5;
    int o = idx & 31;
    float w = 0.f;
    if (k < 81 && o < 25) {
        int a1 = k / 9;
        int a2 = k - a1 * 9;
        const float2* A = F + a1 * GRID_SZ;            // F1
        const float2* B = F + (9 + a2) * GRID_SZ;      // F2
        const float2* C = F + (18 + o) * GRID_SZ;      // Z
        for (int kk = 0; kk < GRID_SZ; ++kk) {
            float2 a = A[kk], b = B[kk], c = C[kk];
            float pr = a.x * b.x - a.y * b.y;          // Re(AB)
            float pi = a.x * b.y + a.y * b.x;          // Im(AB)
            w += pr * c.x + pi * c.y;                  // Re(conj(AB)*C)
        }
        w *= (1.0f / 1024.0f);
    }
    Wt[idx] = w;
}

// ---------------------------------------------------------------------------
// Main pass: out[n,o] = sum_k P[n,k] * W[k,o],  P[n,k] = in1[n,k/9]*in2[n,k%9]
// Block = 8 waves x 16 rows = 128 rows.  Per wave: 16x32 output tile,
// K = 84 in 21 chunks of 4 via V_WMMA_F32_16X16X4_F32.
//
// P is materialized in LDS up-front (no divergence/division in the WMMA loop);
// W is stored chunk-transposed [c][n][kk] so A and B fragments are each one
// aligned ds_load_b64.
// ---------------------------------------------------------------------------
__global__ __launch_bounds__(256) void gaunt_main(const float* __restrict__ in1,
                                                  const float* __restrict__ in2,
                                                  const float* __restrict__ Wt,
                                                  float* __restrict__ out) {
    __shared__ float s1[128 * 9];
    __shared__ float s2[128 * 9];
    __shared__ float sWt[KPAD * OPAD];     // [c][n][kk] = ((c*32+n)*4 + kk)
    __shared__ float sP[128 * KPAD];       // [row][k], row stride 84 floats (8B aligned)

    int tid = threadIdx.x;
    int rowbase = blockIdx.x * 128;

    // Stage inputs + weights into LDS (coalesced).
    for (int i = tid; i < 128 * 9; i += 256) {
        s1[i] = in1[rowbase * 9 + i];
        s2[i] = in2[rowbase * 9 + i];
    }
    for (int i = tid; i < KPAD * OPAD; i += 256) {
        int k = i >> 5;                    // 0..83
        int o = i & 31;
        sWt[((k >> 2) * 32 + o) * 4 + (k & 3)] = Wt[i];
    }
    __syncthreads();

    // Materialize P tile: 2 threads per row, split a2 by parity. No division.
    {
        int row = tid >> 1;
        int h   = tid & 1;
        const float* x1 = &s1[row * 9];
        const float* x2 = &s2[row * 9];
        float* pr = &sP[row * KPAD];
        for (int a1 = 0; a1 < 9; ++a1) {
            float x = x1[a1];
            for (int a2 = h; a2 < 9; a2 += 2)
                pr[a1 * 9 + a2] = x * x2[a2];
        }
        if (h == 0) { pr[81] = 0.f; pr[82] = 0.f; pr[83] = 0.f; }
    }
    __syncthreads();

    int lane = tid & 31;
    int wave = tid >> 5;
    int m    = lane & 15;                  // M row within tile
    int kb   = (lane >> 4) << 1;           // 0 for lanes 0-15, 2 for lanes 16-31
    int nlo  = lane & 15;                  // N column within half-tile

    const float2* pRow = (const float2*)&sP[(wave * 16 + m) * KPAD + kb];
    const float2* wB0  = (const float2*)&sWt[nlo * 4 + kb];
    const float2* wB1  = (const float2*)&sWt[(16 + nlo) * 4 + kb];

    v8f c0 = {};
    v8f c1 = {};

    #pragma unroll 3
    for (int c = 0; c < 21; ++c) {
        float2 af = pRow[c * 2];           // P[m, 4c+kb], P[m, 4c+kb+1]
        float2 b0f = wB0[c * 64];          // W[4c+kb..+1, nlo]
        float2 b1f = wB1[c * 64];          // W[4c+kb..+1, 16+nlo]
        v2f a, b0, b1;
        a.x = af.x;  a.y = af.y;
        b0.x = b0f.x; b0.y = b0f.y;
        b1.x = b1f.x; b1.y = b1f.y;
        c0 = __builtin_amdgcn_wmma_f32_16x16x4_f32(false, a, false, b0,
                                                   (short)0, c0, false, false);
        c1 = __builtin_amdgcn_wmma_f32_16x16x4_f32(false, a, false, b1,
                                                   (short)0, c1, false, false);
    }

    // D layout: VGPR r -> M = r (+8 for lanes 16-31), N = nlo (tile0) / 16+nlo (tile1)
    int row = rowbase + wave * 16 + ((lane < 16) ? 0 : 8);
    for (int r = 0; r < 8; ++r) {
        out[(row + r) * 25 + nlo] = c0[r];
        if (nlo < 9) out[(row + r) * 25 + 16 + nlo] = c1[r];
    }
}

// ---------------------------------------------------------------------------
extern "C" void kernel_launch(void* const* d_in, const int* in_sizes, int n_in,
                              void* d_out, int out_size, void* d_ws, size_t ws_size,
                              hipStream_t stream) {
    const float*  in1 = (const float*)d_in[0];
    const float*  in2 = (const float*)d_in[1];
    const float2* y1  = (const float2*)d_in[2];
    const float2* y2  = (const float2*)d_in[3];
    const float2* zg  = (const float2*)d_in[4];
    float* out = (float*)d_out;

    char* ws = (char*)d_ws;
    float2* tmp = (float2*)ws;                                   // 43*1024 c64 = 344 KB
    float2* F   = (float2*)(ws + (size_t)NGRIDS * GRID_SZ * 8);  // 43*1024 c64 = 344 KB
    float*  Wt  = (float*)(ws + (size_t)NGRIDS * GRID_SZ * 16);  // 84*32 f32   = 10.5 KB

    int N = in_sizes[0] / 9;

    int dft_blocks = (NGRIDS * GRID_SZ + 255) / 256;   // 172
    dft_stage_p<<<dft_blocks, 256, 0, stream>>>(y1, y2, zg, tmp);
    dft_stage_t<<<dft_blocks, 256, 0, stream>>>(tmp, F);
    build_w<<<(KPAD * OPAD + 255) / 256, 256, 0, stream>>>(F, Wt);
    gaunt_main<<<N / 128, 256, 0, stream>>>(in1, in2, Wt, out);
}